// SetEncoder_87497073754841
// MI455X (gfx1250) — compile-verified
//
#include <hip/hip_runtime.h>
#include <cstdint>

#define D 256
#define NF 9
#define BF 3
#define BOND_ROWS 13          // sum(BOND_DIMS)
#define MAX_E 512             // slots per batch row (reference constant)
#define SLOTS_PER_BLOCK 4     // 64 lanes per slot, 256-thread block = 8 wave32

typedef float v4f __attribute__((ext_vector_type(4)));

__device__ __forceinline__ void nt_store_v4(float* p, v4f x) {
    __builtin_nontemporal_store(x, (v4f*)p);
}
__device__ __forceinline__ v4f ld_v4(const float* p) {
    return *(const v4f*)p;
}

// ---------------------------------------------------------------------------
// atom_h[a, :] = sum_f atom_emb[atom_input[a,f] + atom_offsets[f], :]
// 64 lanes per atom, one float4 per lane. Regular stores: we WANT atom_h in L2
// (128 MB < 192 MB global L2) for the gather phase.
// ---------------------------------------------------------------------------
__global__ __launch_bounds__(256) void atom_h_kernel(const int* __restrict__ atom_input,
                                                     const float* __restrict__ atom_emb,
                                                     const int* __restrict__ atom_offsets,
                                                     float* __restrict__ atom_h,
                                                     int n_atoms) {
    int gid  = blockIdx.x * blockDim.x + threadIdx.x;
    int atom = gid >> 6;
    int c    = (gid & 63) << 2;
    if (atom >= n_atoms) return;
    v4f acc = {0.f, 0.f, 0.f, 0.f};
#pragma unroll
    for (int f = 0; f < NF; ++f) {
        int row = atom_input[atom * NF + f] + atom_offsets[f];
        acc += ld_v4(atom_emb + (long)row * D + c);
    }
    *(v4f*)(atom_h + (long)atom * D + c) = acc;
}

// ---------------------------------------------------------------------------
// Recover per-batch edge-range starts from dest_idx (strictly increasing;
// batch i occupies slots i*MAX_E + [0, n_i)). prefix[i] = first edge of batch
// i; prefix[B] = n_edges. Init to n_edges so (impossible) empty batches fail
// safe to "write zeros".
// ---------------------------------------------------------------------------
__global__ void prefix_init_kernel(int* __restrict__ prefix, int B, int n_edges) {
    int i = blockIdx.x * blockDim.x + threadIdx.x;
    if (i <= B) prefix[i] = n_edges;
}

__global__ void prefix_scan_kernel(const int* __restrict__ dest_idx,
                                   int* __restrict__ prefix, int n_edges) {
    int k = blockIdx.x * blockDim.x + threadIdx.x;
    if (k < n_edges) {
        int d = dest_idx[k];
        if ((d & (MAX_E - 1)) == 0) prefix[d >> 9] = k;   // j == 0 -> batch start
    }
}

// ---------------------------------------------------------------------------
// Stage bond_emb (13x256 f32 = 13 KB) into LDS with CDNA5 async copies.
// ---------------------------------------------------------------------------
__device__ __forceinline__ void stage_bond_lds(float* lds_bond, const float* __restrict__ bond_emb) {
    const int total_f4 = (BOND_ROWS * D) / 4;                 // 832
    unsigned lds_base  = (unsigned)(size_t)lds_bond;          // low 32 bits = LDS byte addr
    for (int i = threadIdx.x; i < total_f4; i += blockDim.x) {
        unsigned lds_a = lds_base + (unsigned)i * 16u;
        unsigned long long ga = (unsigned long long)(size_t)bond_emb + (unsigned long long)i * 16ull;
        asm volatile("global_load_async_to_lds_b128 %0, %1, off"
                     :: "v"(lds_a), "v"(ga) : "memory");
    }
    asm volatile("s_wait_asynccnt 0" ::: "memory");
    __syncthreads();
}

// ---------------------------------------------------------------------------
// Single-pass output: one 64-lane group per output slot s = i*MAX_E + j.
// valid (j < n_i): h = atom_h[u] + atom_h[v] + bond rows (LDS); else zeros.
// Every output byte written exactly once, non-temporally. Mask written here
// too (1.0 / 0.0), matching the reference mask.
// ---------------------------------------------------------------------------
__global__ __launch_bounds__(256) void slot_kernel(const float* __restrict__ atom_h,
                                                   const int* __restrict__ bond_input,
                                                   const int* __restrict__ u,
                                                   const int* __restrict__ v,
                                                   const int* __restrict__ prefix,
                                                   const float* __restrict__ bond_emb,
                                                   const int* __restrict__ bond_offsets,
                                                   float* __restrict__ out,
                                                   float* __restrict__ mask_out,
                                                   long rows) {
    __shared__ float lds_bond[BOND_ROWS * D];   // 13,312 bytes
    stage_bond_lds(lds_bond, bond_emb);

    int  s_local = threadIdx.x >> 6;
    int  c       = (threadIdx.x & 63) << 2;
    long s       = (long)blockIdx.x * SLOTS_PER_BLOCK + s_local;
    if (s >= rows) return;

    int i  = (int)(s >> 9);          // s / MAX_E
    int j  = (int)(s & (MAX_E - 1)); // s % MAX_E
    int p0 = prefix[i];
    int p1 = prefix[i + 1];

    v4f  acc   = {0.f, 0.f, 0.f, 0.f};
    bool valid = j < (p1 - p0);
    if (valid) {
        int k  = p0 + j;
        int k2 = 2 * k;
        int u1 = u[k2];
        int v1 = v[k2];
        acc = ld_v4(atom_h + (long)u1 * D + c) + ld_v4(atom_h + (long)v1 * D + c);
#pragma unroll
        for (int f = 0; f < BF; ++f) {
            int row = bond_input[k2 * BF + f] + bond_offsets[f];
            acc += *(const v4f*)(lds_bond + row * D + c);   // ds_load_b128
        }
    }
    nt_store_v4(out + s * D + c, acc);
    if (c == 0) __builtin_nontemporal_store(valid ? 1.0f : 0.0f, mask_out + s);
}

// ---------------------------------------------------------------------------
// Fallback path (ws too small for atom_h): NT-zero everything, then per-edge
// fused recompute from the L2-resident atom_emb table + scatter.
// ---------------------------------------------------------------------------
__global__ __launch_bounds__(256) void zero_kernel(float* __restrict__ out, long n4) {
    long i = (long)blockIdx.x * blockDim.x + threadIdx.x;
    if (i < n4) {
        v4f z = {0.f, 0.f, 0.f, 0.f};
        nt_store_v4(out + i * 4, z);
    }
}

__global__ __launch_bounds__(256) void edge_fused_kernel(const int* __restrict__ atom_input,
                                                         const float* __restrict__ atom_emb,
                                                         const int* __restrict__ atom_offsets,
                                                         const int* __restrict__ bond_input,
                                                         const int* __restrict__ u,
                                                         const int* __restrict__ v,
                                                         const int* __restrict__ dest_idx,
                                                         const float* __restrict__ bond_emb,
                                                         const int* __restrict__ bond_offsets,
                                                         float* __restrict__ out,
                                                         float* __restrict__ mask_out,
                                                         int n_edges) {
    __shared__ float lds_bond[BOND_ROWS * D];
    stage_bond_lds(lds_bond, bond_emb);

    int e_local = threadIdx.x >> 6;
    int c       = (threadIdx.x & 63) << 2;
    int k       = blockIdx.x * SLOTS_PER_BLOCK + e_local;
    if (k >= n_edges) return;

    int k2 = 2 * k;
    int u1 = u[k2];
    int v1 = v[k2];

    v4f acc = {0.f, 0.f, 0.f, 0.f};
#pragma unroll
    for (int f = 0; f < NF; ++f) {
        int off = atom_offsets[f];
        acc += ld_v4(atom_emb + (long)(atom_input[u1 * NF + f] + off) * D + c);
        acc += ld_v4(atom_emb + (long)(atom_input[v1 * NF + f] + off) * D + c);
    }
#pragma unroll
    for (int f = 0; f < BF; ++f) {
        int row = bond_input[k2 * BF + f] + bond_offsets[f];
        acc += *(const v4f*)(lds_bond + row * D + c);
    }
    int dst = dest_idx[k];
    nt_store_v4(out + (long)dst * D + c, acc);
    if (c == 0) __builtin_nontemporal_store(1.0f, mask_out + dst);
}

extern "C" void kernel_launch(void* const* d_in, const int* in_sizes, int n_in,
                              void* d_out, int out_size, void* d_ws, size_t ws_size,
                              hipStream_t stream) {
    const int*   atom_input   = (const int*)d_in[0];
    const int*   bond_input   = (const int*)d_in[1];
    const int*   u            = (const int*)d_in[2];
    const int*   v            = (const int*)d_in[3];
    const int*   dest_idx     = (const int*)d_in[4];
    /* d_in[5] = mask (unused; reconstructed exactly from dest ranges) */
    const float* atom_emb     = (const float*)d_in[6];
    const float* bond_emb     = (const float*)d_in[7];
    const int*   atom_offsets = (const int*)d_in[8];
    const int*   bond_offsets = (const int*)d_in[9];

    int  n_atoms = in_sizes[0] / NF;
    int  n_edges = in_sizes[4];
    long rows    = (long)out_size / (D + 1);   // B*MAX_E (payload + mask tail)
    int  B       = (int)(rows / MAX_E);

    float* out      = (float*)d_out;
    float* mask_out = out + rows * (long)D;

    size_t need_atom_h = (size_t)n_atoms * D * sizeof(float);
    size_t prefix_off  = (need_atom_h + 127) & ~(size_t)127;
    size_t need_total  = prefix_off + (size_t)(B + 1) * sizeof(int);

    if (ws_size >= need_total) {
        float* atom_h = (float*)d_ws;
        int*   prefix = (int*)((char*)d_ws + prefix_off);

        // 1) per-batch edge-range starts from dest_idx
        prefix_init_kernel<<<(B + 256) / 256, 256, 0, stream>>>(prefix, B, n_edges);
        prefix_scan_kernel<<<(n_edges + 255) / 256, 256, 0, stream>>>(dest_idx, prefix, n_edges);

        // 2) materialize atom_h (128 MB -> stays L2-resident)
        long t = (long)n_atoms * 64;
        atom_h_kernel<<<(unsigned)((t + 255) / 256), 256, 0, stream>>>(
            atom_input, atom_emb, atom_offsets, atom_h, n_atoms);

        // 3) single-pass gather + NT write of every output slot (data + mask)
        long nblk = (rows + SLOTS_PER_BLOCK - 1) / SLOTS_PER_BLOCK;
        slot_kernel<<<(unsigned)nblk, 256, 0, stream>>>(
            atom_h, bond_input, u, v, prefix, bond_emb, bond_offsets,
            out, mask_out, rows);
    } else {
        long n4 = (long)out_size / 4;
        zero_kernel<<<(unsigned)((n4 + 255) / 256), 256, 0, stream>>>(out, n4);
        edge_fused_kernel<<<(unsigned)((n_edges + SLOTS_PER_BLOCK - 1) / SLOTS_PER_BLOCK),
                            256, 0, stream>>>(
            atom_input, atom_emb, atom_offsets, bond_input, u, v, dest_idx,
            bond_emb, bond_offsets, out, mask_out, n_edges);
    }
}